// SpecAugmenter_51694226375336
// MI455X (gfx1250) — compile-verified
//
#include <hip/hip_runtime.h>
#include <stdint.h>

// ---------------------------------------------------------------------------
// SpecAugment for MI455X (gfx1250).
// Memory-bound: 164MB traffic -> ~7us at 23.3 TB/s HBM (fits in 192MB L2).
// Main kernel streams float4 through an async global->LDS pipeline
// (GLOBAL_LOAD_ASYNC_TO_LDS_B128 + s_wait_asynccnt), masks, stores B128.
// RNG replicates JAX threefry2x32, partitionable mode (JAX >= 0.4.36 default).
// ---------------------------------------------------------------------------

namespace {
constexpr int kB      = 64;
constexpr int kM      = 80;
constexpr int kT      = 4000;
constexpr int kT4     = kT / 4;          // 1000 float4 per row
constexpr int kRow4   = kM * kT4;        // 80000 float4 per batch sample
constexpr int kN4     = kB * kRow4;      // 5,120,000 float4 total
constexpr int kBlock  = 256;             // 8 waves (wave32)
constexpr int kStages = 8;
constexpr int kPerBlk = kBlock * kStages; // 2048 float4 per block
constexpr int kGrid   = kN4 / kPerBlk;    // 2500 blocks (exact)
constexpr int kDepth  = 4;                // async pipeline depth (LDS buffers)

// workspace layout (bytes)
constexpr size_t WS_FMASK = 0;      // 64 * 4  u32  (80-bit freq mask, padded)
constexpr size_t WS_TPAIR = 1024;   // 64 * 5 * 2 i32 (t0, t0+t)
constexpr size_t WS_TMASK = 4096;   // 64 * 128 u32 (4000-bit time mask, padded)
} // namespace

// ------------------------------ threefry2x32 -------------------------------
__device__ __forceinline__ uint32_t rotl32(uint32_t x, int n) {
  return (x << n) | (x >> (32 - n));
}

__device__ __forceinline__ void tf2x32(uint32_t k0, uint32_t k1,
                                       uint32_t x0, uint32_t x1,
                                       uint32_t& o0, uint32_t& o1) {
  const uint32_t k2 = k0 ^ k1 ^ 0x1BD11BDAu;
  x0 += k0; x1 += k1;
#define SA_R4(a, b, c, d)                                   \
  x0 += x1; x1 = rotl32(x1, a); x1 ^= x0;                   \
  x0 += x1; x1 = rotl32(x1, b); x1 ^= x0;                   \
  x0 += x1; x1 = rotl32(x1, c); x1 ^= x0;                   \
  x0 += x1; x1 = rotl32(x1, d); x1 ^= x0;
  SA_R4(13, 15, 26, 6)
  x0 += k1; x1 += k2 + 1u;
  SA_R4(17, 29, 16, 24)
  x0 += k2; x1 += k0 + 2u;
  SA_R4(13, 15, 26, 6)
  x0 += k0; x1 += k1 + 3u;
  SA_R4(17, 29, 16, 24)
  x0 += k1; x1 += k2 + 4u;
  SA_R4(13, 15, 26, 6)
  x0 += k2; x1 += k0 + 5u;
#undef SA_R4
  o0 = x0; o1 = x1;
}

// partitionable-mode helpers: counter = (hi=0, lo=j)
__device__ __forceinline__ uint32_t tf_bits(uint32_t k0, uint32_t k1, uint32_t j) {
  uint32_t a, b;
  tf2x32(k0, k1, 0u, j, a, b);
  return a ^ b;
}
__device__ __forceinline__ float tf_unif(uint32_t k0, uint32_t k1, uint32_t j) {
  uint32_t bits = tf_bits(k0, k1, j);
  return __uint_as_float((bits >> 9) | 0x3F800000u) - 1.0f;
}
__device__ __forceinline__ int imin(int a, int b) { return a < b ? a : b; }
__device__ __forceinline__ int imax(int a, int b) { return a > b ? a : b; }

// bitmask of [lo,hi) intersected with word [base, base+32)
__device__ __forceinline__ uint32_t range_word(int lo, int hi, int base) {
  int l = imax(lo - base, 0);
  int h = imin(hi - base, 32);
  if (h <= l) return 0u;
  uint32_t mh = (h >= 32) ? 0xFFFFFFFFu : ((1u << h) - 1u);
  uint32_t ml = (l <= 0) ? 0u : ((1u << l) - 1u);
  return mh & ~ml;
}

// ------------------------- kernel 1: per-sample RNG ------------------------
__global__ void sa_setup(const int* __restrict__ lengths,
                         uint32_t* __restrict__ fmask,
                         int* __restrict__ tpair) {
  const int b = threadIdx.x;
  if (b >= kB) return;

  // root key from jax.random.key(42) -> (hi=0, lo=42)
  const uint32_t R0 = 0u, R1 = 42u;
  uint32_t kfw0, kfw1, kfs0, kfs1, ktw0, ktw1, kts0, kts1;
  tf2x32(R0, R1, 0u, 0u, kfw0, kfw1);  // kf_w
  tf2x32(R0, R1, 0u, 1u, kfs0, kfs1);  // kf_s
  tf2x32(R0, R1, 0u, 2u, ktw0, ktw1);  // kt_w
  tf2x32(R0, R1, 0u, 3u, kts0, kts1);  // kt_s

  // randint(kf_w, (B,2), 0, 16): span=16 (pow2) -> lower_bits & 15,
  // lower_bits drawn from second subkey of split(kf_w, 2).
  uint32_t rk0, rk1;
  tf2x32(kfw0, kfw1, 0u, 1u, rk0, rk1);

  uint32_t fm0 = 0u, fm1 = 0u, fm2 = 0u;
#pragma unroll
  for (int j = 0; j < 2; ++j) {
    const uint32_t idx = 2u * (uint32_t)b + (uint32_t)j;
    const int f = (int)(tf_bits(rk0, rk1, idx) & 15u);
    const float uf = tf_unif(kfs0, kfs1, idx);
    const int f0 = (int)floorf(uf * (float)(kM - f + 1));
    fm0 |= range_word(f0, f0 + f, 0);
    fm1 |= range_word(f0, f0 + f, 32);
    fm2 |= range_word(f0, f0 + f, 64);
  }
  fmask[b * 4 + 0] = fm0;
  fmask[b * 4 + 1] = fm1;
  fmask[b * 4 + 2] = fm2;
  fmask[b * 4 + 3] = 0u;

  const int vt = lengths[b];
  int mt = imin(25, (int)((float)vt * 0.05f));
  mt = imin(mt, vt - 1);
  mt = imax(mt, 0);
#pragma unroll
  for (int j = 0; j < 5; ++j) {
    const uint32_t idx = 5u * (uint32_t)b + (uint32_t)j;
    const float ut = tf_unif(ktw0, ktw1, idx);
    int t = (int)floorf(ut * (float)(mt + 1));
    if (vt <= 1) t = 0;
    const float u0 = tf_unif(kts0, kts1, idx);
    const int t0 = (int)floorf(u0 * (float)(vt - t + 1));
    tpair[(b * 5 + j) * 2 + 0] = t0;
    tpair[(b * 5 + j) * 2 + 1] = t0 + t;
  }
}

// --------------------- kernel 2: expand time bitmask -----------------------
__global__ void sa_tmask(const int* __restrict__ tpair,
                         uint32_t* __restrict__ tmask) {
  const int b = blockIdx.x;
  const int w = threadIdx.x;  // 0..127
  uint32_t word = 0u;
  if (w < 125) {
    const int base = w * 32;
#pragma unroll
    for (int j = 0; j < 5; ++j) {
      const int t0 = tpair[(b * 5 + j) * 2 + 0];
      const int te = tpair[(b * 5 + j) * 2 + 1];
      word |= range_word(t0, te, base);
    }
  }
  tmask[b * 128 + w] = word;
}

// -------------------------- async-to-LDS plumbing --------------------------
#define SA_AS1 __attribute__((address_space(1)))
#define SA_AS3 __attribute__((address_space(3)))

typedef __attribute__((ext_vector_type(4))) int sa_v4i;

#if defined(__AMDGCN__) && __has_builtin(__builtin_amdgcn_global_load_async_to_lds_b128)
#define SA_ASYNC 1
#else
#define SA_ASYNC 0
#endif

#if SA_ASYNC
__device__ __forceinline__ void sa_async_ld(const float4* g, float4* l) {
  // signature (probe-confirmed via diagnostic): (v4i AS1*, v4i AS3*, imm, imm)
  __builtin_amdgcn_global_load_async_to_lds_b128(
      (SA_AS1 sa_v4i*)g, (SA_AS3 sa_v4i*)l, 0, 0);
}
#endif

template <int N>
__device__ __forceinline__ void sa_wait_async() {
#if defined(__AMDGCN__)
#if __has_builtin(__builtin_amdgcn_s_wait_asynccnt)
  __builtin_amdgcn_s_wait_asynccnt(N);
#else
  asm volatile("s_wait_asynccnt %0" ::"i"(N) : "memory");
#endif
#endif
}

__device__ __forceinline__ float4 sa_kill(float4 v, uint32_t kill) {
  float4 o;
  o.x = (kill & 1u) ? 0.0f : v.x;
  o.y = (kill & 2u) ? 0.0f : v.y;
  o.z = (kill & 4u) ? 0.0f : v.z;
  o.w = (kill & 8u) ? 0.0f : v.w;
  return o;
}

// ------------------------- kernel 3: apply masks ---------------------------
__global__ __launch_bounds__(kBlock) void sa_apply(
    const float4* __restrict__ in, float4* __restrict__ out,
    const uint32_t* __restrict__ fmask, const uint32_t* __restrict__ tmask) {
  const int tid = threadIdx.x;
  const uint32_t base = (uint32_t)blockIdx.x * kPerBlk;

#if SA_ASYNC
  __shared__ float4 buf[kDepth][kBlock];  // 16KB: depth-4 per-wave pipeline
#pragma unroll
  for (int p = 0; p < kDepth - 1; ++p)
    sa_async_ld(&in[base + p * kBlock + tid], &buf[p][tid]);

#pragma unroll
  for (int s = 0; s < kStages; ++s) {
    if (s + kDepth - 1 < kStages)
      sa_async_ld(&in[base + (s + kDepth - 1) * kBlock + tid],
                  &buf[(s + kDepth - 1) & (kDepth - 1)][tid]);
    const int allow = imin(kStages - 1 - s, kDepth - 1);
    if (allow >= 3)      sa_wait_async<3>();
    else if (allow == 2) sa_wait_async<2>();
    else if (allow == 1) sa_wait_async<1>();
    else                 sa_wait_async<0>();

    const float4 v = buf[s & (kDepth - 1)][tid];
    const uint32_t i = base + (uint32_t)(s * kBlock + tid);
    const uint32_t b = i / (uint32_t)kRow4;
    const uint32_t r = i - b * (uint32_t)kRow4;
    const uint32_t m = r / (uint32_t)kT4;
    const uint32_t c = r - m * (uint32_t)kT4;
    const uint32_t fbit = (fmask[(b << 2) + (m >> 5)] >> (m & 31u)) & 1u;
    const uint32_t tb = c << 2;
    const uint32_t t4 = (tmask[(b << 7) + (tb >> 5)] >> (tb & 31u)) & 0xFu;
    out[i] = sa_kill(v, t4 | (fbit ? 0xFu : 0u));
  }
#else
  // fallback: direct vectorized streaming, skip loads of fully-masked rows
#pragma unroll
  for (int s = 0; s < kStages; ++s) {
    const uint32_t i = base + (uint32_t)(s * kBlock + tid);
    const uint32_t b = i / (uint32_t)kRow4;
    const uint32_t r = i - b * (uint32_t)kRow4;
    const uint32_t m = r / (uint32_t)kT4;
    const uint32_t c = r - m * (uint32_t)kT4;
    const uint32_t fbit = (fmask[(b << 2) + (m >> 5)] >> (m & 31u)) & 1u;
    const uint32_t tb = c << 2;
    const uint32_t t4 = (tmask[(b << 7) + (tb >> 5)] >> (tb & 31u)) & 0xFu;
    float4 v = make_float4(0.0f, 0.0f, 0.0f, 0.0f);
    if (!fbit) v = in[i];  // wave-coherent: lanes share the mel row
    out[i] = sa_kill(v, t4);
  }
#endif
}

// ------------------------------- launcher ----------------------------------
extern "C" void kernel_launch(void* const* d_in, const int* in_sizes, int n_in,
                              void* d_out, int out_size, void* d_ws,
                              size_t ws_size, hipStream_t stream) {
  (void)in_sizes; (void)n_in; (void)out_size; (void)ws_size;
  const float* mel     = (const float*)d_in[0];
  const int*   lengths = (const int*)d_in[1];
  float*       out     = (float*)d_out;
  uint8_t*     ws      = (uint8_t*)d_ws;

  uint32_t* fmask = (uint32_t*)(ws + WS_FMASK);
  int*      tpair = (int*)(ws + WS_TPAIR);
  uint32_t* tmask = (uint32_t*)(ws + WS_TMASK);

  sa_setup<<<dim3(1), dim3(64), 0, stream>>>(lengths, fmask, tpair);
  sa_tmask<<<dim3(kB), dim3(128), 0, stream>>>(tpair, tmask);
  sa_apply<<<dim3(kGrid), dim3(kBlock), 0, stream>>>(
      (const float4*)mel, (float4*)out, fmask, tmask);
}